// Decoder_56427280335511
// MI455X (gfx1250) — compile-verified
//
#include <hip/hip_runtime.h>

#define Hdim 128
#define Gseg 1024
#define PX 132        // X tile pitch (floats): A-frag b64 reads bank-conflict-free
#define WP2 288       // W K-pair row pitch (floats): 288 % 64 == 32 -> lane halves hit disjoint banks
#define TILE_M 128
#define NTHREADS 256  // 8 wave32 waves; each wave owns 16 rows x full 128 cols

typedef __attribute__((ext_vector_type(2))) float v2f;
typedef __attribute__((ext_vector_type(4))) float v4f;
typedef __attribute__((ext_vector_type(8))) float v8f;

__device__ __forceinline__ float ssp_f(float x) {
    // softplus(x) - ln(2), numerically stable
    return fmaxf(x, 0.0f) + log1pf(expf(-fabsf(x))) - 0.69314718055994530942f;
}

__global__ __launch_bounds__(NTHREADS) void zero_kernel(float* __restrict__ p, int n) {
    int i = blockIdx.x * blockDim.x + threadIdx.x;
    if (i < n) p[i] = 0.0f;
}

// Fused: 3x (X@W + b -> LayerNorm -> shifted softplus), then per-row dot with wvec,
// atomically segment-summed into accOut. One 128-row tile per workgroup, all in LDS.
__global__ __launch_bounds__(NTHREADS) void mlp3_pool_kernel(
    const float* __restrict__ Xg, const int* __restrict__ batch,
    const float* __restrict__ Wg, const float* __restrict__ bg,
    const float* __restrict__ gg, const float* __restrict__ btg,
    const float* __restrict__ wvec, float* __restrict__ accOut,
    int R, int N, int isProc)
{
    extern __shared__ float smem[];
    float* sX  = smem;                   // TILE_M * PX
    float* sW  = sX + TILE_M * PX;       // (Hdim/2) * WP2, K-pair interleaved: [k/2][n][2]
    float* sB  = sW + (Hdim / 2) * WP2;  // bias   [128]
    float* sG  = sB + Hdim;              // gain   [128]
    float* sBt = sG + Hdim;              // beta   [128]
    float* sWv = sBt + Hdim;             // w proj [128]

    const int tid  = threadIdx.x;
    const int wave = tid >> 5;
    const int lane = tid & 31;
    const int half = lane >> 4;
    const int l16  = lane & 15;
    const int m0   = wave * 16;
    const int R0   = blockIdx.x * TILE_M;

    // Stage input tile into LDS (streamed once from HBM; non-temporal keeps L2 for weights)
    for (int idx = tid; idx < TILE_M * (Hdim / 4); idx += NTHREADS) {
        int r  = idx >> 5;
        int c4 = (idx & 31) << 2;
        int rowG = R0 + r;
        v4f x4 = {0.f, 0.f, 0.f, 0.f};
        if (rowG < R)
            x4 = __builtin_nontemporal_load((const v4f*)(Xg + (size_t)rowG * Hdim + c4));
        float* d = &sX[r * PX + c4];
        d[0] = x4.x; d[1] = x4.y; d[2] = x4.z; d[3] = x4.w;
    }
    if (tid < Hdim) sWv[tid] = wvec[tid];

    for (int s = 0; s < 3; ++s) {
        if (s) __syncthreads();                  // all waves done reading previous sW
        const float* Ws = Wg + s * Hdim * Hdim;  // stage weights -> LDS (L2-resident)
        for (int idx = tid; idx < Hdim * (Hdim / 4); idx += NTHREADS) {
            int r  = idx >> 5;
            int c4 = (idx & 31) << 2;
            v4f w4 = *(const v4f*)(Ws + r * Hdim + c4);
            // K-pair interleave: (k,n) and (k+1,n) adjacent -> B frag is one ds_load_b64
            float* d = &sW[(r >> 1) * WP2 + (c4 << 1) + (r & 1)];
            d[0] = w4.x; d[2] = w4.y; d[4] = w4.z; d[6] = w4.w;
        }
        if (tid < Hdim) {
            sB[tid]  = bg[s * Hdim + tid];
            sG[tid]  = gg[s * Hdim + tid];
            sBt[tid] = btg[s * Hdim + tid];
        }
        __syncthreads();

        // GEMM: this wave's 16 rows x 128 cols = 8 accumulator tiles (16x16 f32)
        v8f acc[8];
        const v8f vz = {0.f, 0.f, 0.f, 0.f, 0.f, 0.f, 0.f, 0.f};
        #pragma unroll
        for (int t = 0; t < 8; ++t) acc[t] = vz;

        const float* xrow  = &sX[(m0 + l16) * PX];
        const float* wbase = &sW[half * WP2 + 2 * l16];
        #pragma unroll 4
        for (int k = 0; k < Hdim; k += 4) {
            // A frag 16x4: lanes 0-15 hold K={k,k+1}, lanes 16-31 hold K={k+2,k+3}
            v2f a = *(const v2f*)(xrow + k + 2 * half);
            const float* wk = wbase + (k >> 1) * WP2;
            #pragma unroll
            for (int t = 0; t < 8; ++t) {
                v2f bfrag = *(const v2f*)(wk + 32 * t);   // single aligned b64, conflict-free
                acc[t] = __builtin_amdgcn_wmma_f32_16x16x4_f32(
                    false, a, false, bfrag, (short)0, acc[t], false, false);
            }
        }

        // bias + LayerNorm + SSP; C/D layout: VGPR j -> row m0+j (+8 for high half), col t*16+l16
        #pragma unroll
        for (int j = 0; j < 8; ++j) {
            float vals[8];
            float s1 = 0.f, s2 = 0.f;
            #pragma unroll
            for (int t = 0; t < 8; ++t) {
                float v = acc[t][j] + sB[t * 16 + l16];
                vals[t] = v; s1 += v; s2 += v * v;
            }
            #pragma unroll
            for (int off = 1; off < 16; off <<= 1) {   // row lives in one 16-lane half
                s1 += __shfl_xor(s1, off, 32);
                s2 += __shfl_xor(s2, off, 32);
            }
            float mean = s1 * (1.0f / Hdim);
            float var  = s2 * (1.0f / Hdim) - mean * mean;
            float rstd = rsqrtf(var + 1e-5f);
            int rloc = m0 + j + half * 8;
            if (s < 2) {
                float* xo = &sX[rloc * PX];            // rows private to this wave
                #pragma unroll
                for (int t = 0; t < 8; ++t) {
                    int n = t * 16 + l16;
                    float v = (vals[t] - mean) * rstd * sG[n] + sBt[n];
                    xo[n] = ssp_f(v);
                }
            } else {
                // final stage: fuse projection + segment-sum (never write h to HBM)
                float dot = 0.f;
                #pragma unroll
                for (int t = 0; t < 8; ++t) {
                    int n = t * 16 + l16;
                    float v = (vals[t] - mean) * rstd * sG[n] + sBt[n];
                    dot += ssp_f(v) * sWv[n];
                }
                #pragma unroll
                for (int off = 1; off < 16; off <<= 1)
                    dot += __shfl_xor(dot, off, 32);
                if (l16 == 0) {
                    int rowG = R0 + rloc;
                    if (rowG < R) {
                        if (isProc) {
                            int tt = rowG / N;
                            int ii = rowG - tt * N;
                            atomicAdd(&accOut[tt * Gseg + batch[ii]], dot);
                        } else {
                            atomicAdd(&accOut[batch[rowG]], dot);
                        }
                    }
                }
            }
        }
    }
}

__global__ __launch_bounds__(NTHREADS) void combine_kernel(
    float* __restrict__ out, const float* __restrict__ accP,
    const float* __restrict__ accE, const float* __restrict__ bp,
    const float* __restrict__ be, int n)
{
    int i = blockIdx.x * blockDim.x + threadIdx.x;
    if (i < n) out[i] = accP[i] + bp[0] + accE[i & (Gseg - 1)] + be[0];
}

extern "C" void kernel_launch(void* const* d_in, const int* in_sizes, int n_in,
                              void* d_out, int out_size, void* d_ws, size_t ws_size,
                              hipStream_t stream)
{
    (void)n_in; (void)out_size; (void)ws_size;
    const float* x_proc = (const float*)d_in[0];
    const float* x_enc  = (const float*)d_in[1];
    const int*   batch  = (const int*)d_in[2];
    const float* pW  = (const float*)d_in[3];
    const float* pb  = (const float*)d_in[4];
    const float* pg  = (const float*)d_in[5];
    const float* pbt = (const float*)d_in[6];
    const float* eW  = (const float*)d_in[7];
    const float* eb  = (const float*)d_in[8];
    const float* eg  = (const float*)d_in[9];
    const float* ebt = (const float*)d_in[10];
    const float* wp  = (const float*)d_in[11];
    const float* bp  = (const float*)d_in[12];
    const float* we  = (const float*)d_in[13];
    const float* be  = (const float*)d_in[14];
    float* out = (float*)d_out;

    const int N  = in_sizes[2];
    const int T  = in_sizes[0] / (N * Hdim);
    const int Rp = T * N;

    float* accP = (float*)d_ws;        // [T*G]
    float* accE = accP + T * Gseg;     // [G]

    const int smem_bytes =
        (TILE_M * PX + (Hdim / 2) * WP2 + 4 * Hdim) * (int)sizeof(float); // ~140 KB (CDNA5: 320 KB/WGP)
    (void)hipFuncSetAttribute(reinterpret_cast<const void*>(mlp3_pool_kernel),
                              hipFuncAttributeMaxDynamicSharedMemorySize, smem_bytes);

    int nacc = T * Gseg + Gseg;
    zero_kernel<<<(nacc + NTHREADS - 1) / NTHREADS, NTHREADS, 0, stream>>>(accP, nacc);

    mlp3_pool_kernel<<<(Rp + TILE_M - 1) / TILE_M, NTHREADS, smem_bytes, stream>>>(
        x_proc, batch, pW, pb, pg, pbt, wp, accP, Rp, N, 1);
    mlp3_pool_kernel<<<(N + TILE_M - 1) / TILE_M, NTHREADS, smem_bytes, stream>>>(
        x_enc, batch, eW, eb, eg, ebt, we, accE, N, N, 0);

    combine_kernel<<<(T * Gseg + NTHREADS - 1) / NTHREADS, NTHREADS, 0, stream>>>(
        out, accP, accE, bp, be, T * Gseg);
}